// WeightedTensorProduct_62869731279068
// MI455X (gfx1250) — compile-verified
//
#include <hip/hip_runtime.h>

// CDNA5 / gfx1250. Wave32. One 32-lane wave (one workgroup) per batch element z.
//
// Per z:  D(32x16) = W_cat^T(32x128) @ RHS(128x16)  via v_wmma_f32_16x16x4_f32
//   W_cat rows  0.. 31 : w000   -> RHS col 0 = s1      => D col0 = a000
//   W_cat rows 32.. 63 : w011   -> RHS col 2 = s1      => D col2 = a011
//   W_cat rows 64.. 95 : w101   -> RHS col 3..5 = v1   => D col3..5 = b101
//   W_cat rows 96..127 : w110   -> RHS col 1 = t=v1.v2 => D col1 = a110
// then
//   out[z][w]        = 0.125*(s2*a000[w] + a110[w]/sqrt3)
//   out[z][32+3w+k]  = 0.125*(a011[w]*v2[k] + b101[w][k]*s2)
//
// Bandwidth-bound (AI ~0.7 FLOP/B): weights (16 KB/z) streamed once, NT hints,
// unroll 8 => ~24+ loads in flight per wave for latency hiding.

typedef float v2f __attribute__((ext_vector_type(2)));
typedef float v8f __attribute__((ext_vector_type(8)));

__global__ void __launch_bounds__(32)
wtp_wmma_kernel(const float* __restrict__ x1,
                const float* __restrict__ x2,
                const float* __restrict__ weight,
                float* __restrict__ out,
                int n)
{
    const int z = blockIdx.x;
    if (z >= n) return;                    // uniform branch, EXEC stays all-ones

    __shared__ float rhs[128 * 16];        // padded block-diagonal RHS (8 KB)
    __shared__ float dbuf[32 * 6];         // result scatter buffer

    const int lane = threadIdx.x;          // 0..31
    const int lo   = lane & 15;            // N index / M index
    const int hi   = lane >> 4;            // selects K rows {0,1} vs {2,3}

    // ---- per-z small inputs -------------------------------------------------
    const float4 xv = *reinterpret_cast<const float4*>(x2 + (size_t)z * 4);
    const float s2 = xv.x, v2x = xv.y, v2y = xv.z, v2z = xv.w;

    const float* x1z = x1 + (size_t)z * 128;
    const float s1l = x1z[lane];
    const float v1x = x1z[32 + 3 * lane + 0];
    const float v1y = x1z[32 + 3 * lane + 1];
    const float v1z = x1z[32 + 3 * lane + 2];
    const float tl  = v1x * v2x + v1y * v2y + v1z * v2z;

    // ---- stage RHS in LDS: zero cols 0..5 of the 4 rows this lane owns ------
    #pragma unroll
    for (int i = 0; i < 4; ++i) {
        const int row = lane + 32 * i;
        *reinterpret_cast<float4*>(&rhs[row * 16]) = make_float4(0.f, 0.f, 0.f, 0.f);
        rhs[row * 16 + 4] = 0.f;
        rhs[row * 16 + 5] = 0.f;
    }
    // fill the same rows (same lane -> DS in-order, no cross-lane hazard here)
    rhs[(      lane) * 16 + 0] = s1l;   // block0: w000 . s1
    rhs[( 32 + lane) * 16 + 2] = s1l;   // block1: w011 . s1
    rhs[( 64 + lane) * 16 + 3] = v1x;   // block2: w101 . v1
    rhs[( 64 + lane) * 16 + 4] = v1y;
    rhs[( 64 + lane) * 16 + 5] = v1z;
    rhs[( 96 + lane) * 16 + 1] = tl;    // block3: w110 . t
    __syncthreads();                    // single-wave WG: lowers to S_NOP

    // ---- main WMMA loop: 2 M-tiles x 32 K-steps of 16x16x4 f32 --------------
    // A-frag element (M=lo, K=2*hi+v) of tile w0: weight[(u0+2*hi+v)*32 + w0 + lo]
    const float* wl0 = weight + (size_t)z * 4096 + (2 * hi) * 32 + lo;       // w0 = 0
    const float* wl1 = wl0 + 16;                                             // w0 = 16
    const int    lb  = (2 * hi) * 16 + lo;                                   // B-frag LDS idx

    v8f d0 = {0.f, 0.f, 0.f, 0.f, 0.f, 0.f, 0.f, 0.f};
    v8f d1 = {0.f, 0.f, 0.f, 0.f, 0.f, 0.f, 0.f, 0.f};

    #pragma unroll 8
    for (int u0 = 0; u0 < 128; u0 += 4) {
        v2f a0, a1, b;
        // weights streamed once: non-temporal so they do not pollute L2
        a0.x = __builtin_nontemporal_load(wl0 + u0 * 32);
        a0.y = __builtin_nontemporal_load(wl0 + u0 * 32 + 32);
        a1.x = __builtin_nontemporal_load(wl1 + u0 * 32);
        a1.y = __builtin_nontemporal_load(wl1 + u0 * 32 + 32);
        b.x  = rhs[lb + u0 * 16];
        b.y  = rhs[lb + u0 * 16 + 16];

        d0 = __builtin_amdgcn_wmma_f32_16x16x4_f32(false, a0, false, b,
                                                   (short)0, d0, false, false);
        d1 = __builtin_amdgcn_wmma_f32_16x16x4_f32(false, a1, false, b,
                                                   (short)0, d1, false, false);
    }

    // ---- scatter D cols 0..5 to LDS (cross-lane), then combine + store ------
    if (lo < 6) {
        #pragma unroll
        for (int r = 0; r < 8; ++r) {
            dbuf[(     r + 8 * hi) * 6 + lo] = d0[r];   // w = 0..15
            dbuf[(16 + r + 8 * hi) * 6 + lo] = d1[r];   // w = 16..31
        }
    }
    __syncthreads();                    // S_NOP again (ordering for compiler)

    const float a000 = dbuf[lane * 6 + 0];
    const float a110 = dbuf[lane * 6 + 1];
    const float a011 = dbuf[lane * 6 + 2];
    const float b101x = dbuf[lane * 6 + 3];
    const float b101y = dbuf[lane * 6 + 4];
    const float b101z = dbuf[lane * 6 + 5];

    const float INV_SQRT3 = 0.57735026918962576f;
    float* outz = out + (size_t)z * 128;
    outz[lane]              = 0.125f * (s2 * a000 + INV_SQRT3 * a110);
    outz[32 + 3 * lane + 0] = 0.125f * (a011 * v2x + b101x * s2);
    outz[32 + 3 * lane + 1] = 0.125f * (a011 * v2y + b101y * s2);
    outz[32 + 3 * lane + 2] = 0.125f * (a011 * v2z + b101z * s2);
}

extern "C" void kernel_launch(void* const* d_in, const int* in_sizes, int n_in,
                              void* d_out, int out_size, void* d_ws, size_t ws_size,
                              hipStream_t stream)
{
    const float* x1 = (const float*)d_in[0];   // (N, 128)
    const float* x2 = (const float*)d_in[1];   // (N, 4)
    const float* w  = (const float*)d_in[2];   // (N, 4096)
    float* out = (float*)d_out;                // (N, 128)
    const int n = in_sizes[1] / 4;             // N

    wtp_wmma_kernel<<<dim3(n), dim3(32), 0, stream>>>(x1, x2, w, out, n);
}